// MemKDMClassModel_74663711473735
// MI455X (gfx1250) — compile-verified
//
#include <hip/hip_runtime.h>

// MemKDM: B=512 batches, N=1024 neighbors, D=256 features, 100 classes.
// HBM-bound (512 MB x_neigh read, AI ~0.5 flop/byte) -> single pass, f32.
// Dot products via V_WMMA_F32_16X16X4_F32 (f32 WMMA), norms piggy-backed on
// the A-matrix loads, class scatter via LDS f32 atomics.

typedef float v2f __attribute__((ext_vector_type(2)));
typedef float v8f __attribute__((ext_vector_type(8)));

#define B_SZ 512
#define N_SZ 1024
#define D_SZ 256
#define DIMY 100

__global__ __launch_bounds__(256) void memkdm_kernel(
    const float* __restrict__ x_enc,     // [B, D]
    const float* __restrict__ x_neigh,   // [B, N, D]
    const int*   __restrict__ y_neigh,   // [B, N]
    const float* __restrict__ sigma,     // [1]
    float*       __restrict__ out)       // [B, DIMY]
{
    __shared__ float s_xe[D_SZ];   // x_enc row for this batch
    __shared__ float s_probs[DIMY];
    __shared__ float s_sum;
    __shared__ float s_xe2;

    const int b    = blockIdx.x;
    const int tid  = threadIdx.x;
    const int lane = tid & 31;
    const int wave = tid >> 5;

    if (tid < DIMY) s_probs[tid] = 0.0f;
    if (tid == 0) { s_sum = 0.0f; s_xe2 = 0.0f; }
    __syncthreads();

    // Stage x_enc[b] into LDS and reduce its squared norm (256 threads == D).
    {
        float v = x_enc[(size_t)b * D_SZ + tid];
        s_xe[tid] = v;
        atomicAdd(&s_xe2, v * v);
    }
    __syncthreads();

    const float xe2    = s_xe2;
    const float sig    = sigma[0];
    const float inv2s2 = 1.0f / (2.0f * sig * sig);

    // A-matrix (16x4 f32) lane layout: lanes 0-15 hold K=k0,k0+1 of row M=lane;
    // lanes 16-31 hold K=k0+2,k0+3 of row M=lane-16. B (4x16) mirrors it; all
    // 16 B columns carry the same x_enc chunk so C columns replicate the dots.
    const int klo  = (lane >= 16) ? 2 : 0;
    const int mrow = lane & 15;

    float lsum = 0.0f;

    // Each wave owns 128 neighbors: 8 tiles of 16 rows.
    for (int t = 0; t < 8; ++t) {
        const int n0 = wave * 128 + t * 16;
        const float* pa =
            x_neigh + ((size_t)b * N_SZ + (size_t)(n0 + mrow)) * D_SZ + klo;

        v8f  c  = {};     // f32 accumulator (8 VGPRs)
        float sq = 0.0f;  // partial ||x_neigh row||^2 from this lane's A elems

        #pragma unroll 8
        for (int k0 = 0; k0 < D_SZ; k0 += 4) {
            v2f a = *(const v2f*)(pa + k0);              // global_load_b64
            v2f bb = *(const v2f*)(&s_xe[k0 + klo]);     // ds_load_b64
            sq = fmaf(a.x, a.x, sq);
            sq = fmaf(a.y, a.y, sq);
            // D = A(16x4) * B(4x16) + C, f32
            c = __builtin_amdgcn_wmma_f32_16x16x4_f32(
                    false, a, false, bb, (short)0, c, false, false);
        }

        // Combine half-row norm partials: lane L ends with ||row (L&15)||^2.
        float nrm = sq + __shfl_xor(sq, 16, 32);

        // C layout: lanes 0-15 hold rows 0-7 in c[0..7]; lanes 16-31 rows 8-15.
        // Columns are identical, so pick lane v for row v and lane 24+v for
        // row 8+v  (each of those lanes holds exactly that row's norm too).
        #pragma unroll
        for (int v = 0; v < 8; ++v) {
            if (lane == v || lane == 24 + v) {
                const int   row = (lane < 16) ? v : (8 + v);
                const float dot = c[v];
                float d2  = xe2 + nrm - 2.0f * dot;
                d2 = fmaxf(d2, 0.0f);
                float ink = __expf(-d2 * inv2s2);  // underflows exactly like ref
                float val = ink * ink;             // in_k^2
                const int n = n0 + row;
                int y = y_neigh[(size_t)b * N_SZ + n];
                if (y < 0) y = 0;
                if (y >= DIMY) y = DIMY - 1;
                atomicAdd(&s_probs[y], val);       // ds_add_f32
                lsum += val;
            }
        }
    }

    atomicAdd(&s_sum, lsum);
    __syncthreads();

    // probs[b, i] = class_sum_i / max(total, 1e-12)
    const float inv = 1.0f / fmaxf(s_sum, 1e-12f);
    for (int i = tid; i < DIMY; i += 256) {
        out[(size_t)b * DIMY + i] = s_probs[i] * inv;
    }
}

extern "C" void kernel_launch(void* const* d_in, const int* in_sizes, int n_in,
                              void* d_out, int out_size, void* d_ws, size_t ws_size,
                              hipStream_t stream) {
    const float* x_enc   = (const float*)d_in[0];
    const float* x_neigh = (const float*)d_in[1];
    const int*   y_neigh = (const int*)d_in[2];
    const float* sigma   = (const float*)d_in[3];
    float*       out     = (float*)d_out;

    memkdm_kernel<<<dim3(B_SZ), dim3(256), 0, stream>>>(
        x_enc, x_neigh, y_neigh, sigma, out);
}